// XGate_56573309222983
// MI455X (gfx1250) — compile-verified
//
#include <hip/hip_runtime.h>

// CDNA5 / gfx1250 — generalized Pauli-X (cyclic shift) gate on qudit `index`
// of a (4^12, 8) f32 state: out[a,j,c] = sum_i M[j,i] * x[a,i,c]
// with x viewed as (left, 4, R), R = right*batch contiguous.
//
// Bandwidth-bound: 1 GB traffic -> ~43 us @ 23.3 TB/s. Contraction done with
// chained V_WMMA_F32_16X16X4_F32 (exact f32, K=4==D); block-row-placed A
// matrices give 100% utilization of the 16x16 accumulator (4 a-blocks x 4 j
// rows x 16 columns = 256 outputs per 4-WMMA chain).
//
// This round: wave-uniform tile coords forced into SGPRs (readfirstlane) and
// all per-lane addressing reduced to one reusable 32-bit element offset so
// loads/stores use the SADDR + 32-bit-VGPR-offset global addressing mode
// instead of per-address v_add_nc_u64 chains. Fallback path rewritten in
// 32-bit unsigned math (T < 2^31 here) to kill the u64 soft-division bloat.

typedef __attribute__((ext_vector_type(2))) float v2f;
typedef __attribute__((ext_vector_type(8))) float v8f;

#define WAVES_PER_BLOCK 8
#define BLOCK_THREADS (WAVES_PER_BLOCK * 32)

#if defined(__gfx1250__) && __has_builtin(__builtin_amdgcn_wmma_f32_16x16x4_f32)
#define HAVE_WMMA_F32X4 1
#else
#define HAVE_WMMA_F32X4 0
#endif

__global__ __launch_bounds__(BLOCK_THREADS)
void xgate_wmma_kernel(const float* __restrict__ x,
                       const float* __restrict__ M,
                       const int*   __restrict__ indexPtr,
                       float*       __restrict__ out,
                       unsigned T,    // total elements of x / out (< 2^31)
                       int Dd)        // qudit dimension (from M's size)
{
    const int index = indexPtr[0];                 // uniform scalar load
    unsigned left = 1;
    for (int t = 0; t < index; ++t) left *= (unsigned)Dd;
    const unsigned R = T / (left * (unsigned)Dd);  // contiguous row length (right*batch)

    const int lane = threadIdx.x & 31;

#if HAVE_WMMA_F32X4
    if ((Dd == 4) && ((left & 3u) == 0u) && ((R & 15u) == 0u)) {
        const unsigned tiles  = T >> 8;            // 256 outputs per 16x16 D tile
        const unsigned ctiles = R >> 4;            // column tiles per block-row

        // Force the wave id into an SGPR so tile coords (t, g, c0) stay scalar
        // and memory ops use SADDR + 32-bit VGPR offset addressing.
        const unsigned waveId = (unsigned)__builtin_amdgcn_readfirstlane(
            (int)(blockIdx.x * WAVES_PER_BLOCK + (threadIdx.x >> 5)));
        const unsigned nWaves = gridDim.x * WAVES_PER_BLOCK;

        // ---- per-lane WMMA layout constants (wave32) ----
        // A (16x4 f32): lanes 0-15: row=lane, K={0,1}; lanes 16-31: row=lane-16, K={2,3}
        // B (4x16 f32): lanes 0-15: K={0,1}, N=lane;  lanes 16-31: K={2,3}, N=lane-16
        // D (16x16 f32): VGPR r -> lanes 0-15: row r; lanes 16-31: row r+8; N=lane&15
        const int n     = lane & 15;               // column within 16-wide tile
        const int hi    = lane >> 4;               // half-wave select
        const int iPair = hi * 2;                  // first K-row held by this half

        // Block-row-placed A matrices: A_q[row,i] = (row>>2==q) ? M[row&3,i] : 0
        v2f Amat[4];
        {
            const int j    = lane & 3;
            const int qOfM = (lane & 15) >> 2;
            const float m0 = M[j * 4 + iPair];
            const float m1 = M[j * 4 + iPair + 1];
#pragma unroll
            for (int q = 0; q < 4; ++q) {
                Amat[q].x = (qOfM == q) ? m0 : 0.0f;
                Amat[q].y = (qOfM == q) ? m1 : 0.0f;
            }
        }

        // Reusable per-lane 32-bit element offsets (byte offsets < 2^31)
        const unsigned loadOff  = (unsigned)iPair * R + (unsigned)n;       // B row iPair
        const unsigned loadOff2 = loadOff + R;                             // B row iPair+1
        const unsigned storeOff = (unsigned)(hi * 8) * R + (unsigned)n;    // D row base

        for (unsigned t = waveId; t < tiles; t += nWaves) {   // t wave-uniform -> EXEC all-ones
            const unsigned g  = t / ctiles;                   // group of 4 "a" blocks
            const unsigned c0 = (t - g * ctiles) << 4;

            v8f acc = {};
#pragma unroll
            for (int q = 0; q < 4; ++q) {
                // scalar (SGPR) base: block a = 4g+q, column c0
                const float* bp = x + (size_t)((g * 4u + (unsigned)q) * 4u * R + c0);
                v2f bq;
                bq.x = bp[loadOff];
                bq.y = bp[loadOff2];
                acc = __builtin_amdgcn_wmma_f32_16x16x4_f32(
                    /*neg_a=*/false, Amat[q], /*neg_b=*/false, bq,
                    /*c_mod=*/(short)0, acc, /*reuse_a=*/false, /*reuse_b=*/false);
            }

            // Scatter the 16x16 accumulator; row mr = r + 8*hi maps to
            // out offset g*16R + mr*R + c0 + n = (scalar base) + storeOff + r*R
            float* op = out + (size_t)(g * 16u * R + c0);
#pragma unroll
            for (int r = 0; r < 8; ++r) {
                op[storeOff + (unsigned)r * R] = acc[r];
            }
        }
        return;
    }
#endif

    // ---- generic fallback (any D / index), 32-bit unsigned math ----
    {
        const unsigned tid      = blockIdx.x * BLOCK_THREADS + threadIdx.x;
        const unsigned nthreads = gridDim.x * BLOCK_THREADS;
        const unsigned DR       = (unsigned)Dd * R;
        for (unsigned f = tid; f < T; f += nthreads) {
            const unsigned a   = f / DR;
            const unsigned rem = f - a * DR;
            const unsigned j   = rem / R;
            const unsigned c   = rem - j * R;
            const float* xp = x + (size_t)a * DR + c;
            float s = 0.0f;
            for (int i = 0; i < Dd; ++i)
                s = fmaf(M[j * (unsigned)Dd + i], xp[(size_t)i * R], s);
            out[f] = s;
        }
    }
}

extern "C" void kernel_launch(void* const* d_in, const int* in_sizes, int n_in,
                              void* d_out, int out_size, void* d_ws, size_t ws_size,
                              hipStream_t stream) {
    const float* x   = (const float*)d_in[0];
    const float* M   = (const float*)d_in[1];
    const int*   idx = (const int*)d_in[2];
    float*       out = (float*)d_out;

    const unsigned T = (unsigned)in_sizes[0];
    int Dd = 1;
    while ((long long)Dd * (long long)Dd < (long long)in_sizes[1]) ++Dd;  // D = sqrt(|M|)

    long long tiles  = (long long)(T >> 8);
    long long blocks = (tiles + WAVES_PER_BLOCK - 1) / WAVES_PER_BLOCK;
    if (blocks < 1) blocks = 1;
    if (blocks > (1LL << 20)) blocks = (1LL << 20);   // grid-stride covers the rest

    xgate_wmma_kernel<<<(int)blocks, BLOCK_THREADS, 0, stream>>>(x, M, idx, out, T, Dd);
}